// GATModel_13073880449912
// MI455X (gfx1250) — compile-verified
//
#include <hip/hip_runtime.h>
#include <hip/hip_bf16.h>
#include <math.h>

typedef float v2f __attribute__((ext_vector_type(2)));
typedef float v8f __attribute__((ext_vector_type(8)));

#define NEG_SLOPE 0.2f
#define EPS_ 1e-16f

static inline int cdiv(long a, long b) { return (int)((a + b - 1) / b); }

// ---------------------------------------------------------------------------
// generic fill
// ---------------------------------------------------------------------------
__global__ void fill_kernel(float* __restrict__ p, float v, size_t n) {
    size_t i = (size_t)blockIdx.x * blockDim.x + threadIdx.x;
    size_t stride = (size_t)gridDim.x * blockDim.x;
    for (; i < n; i += stride) p[i] = v;
}

// ---------------------------------------------------------------------------
// f32 WMMA GEMM: C[M,NC] = A[M,K] * B[K,NC], K/NC compile-time.
// Each wave owns one 16-row tile and CPW consecutive 16-col tiles
// (A loaded once per k-step, reused across CPW WMMAs).
// Lane layout per CDNA5 ISA 7.12.2:
//   A 16x4: lanes 0-15 hold K=0,1 (v0,v1); lanes 16-31 hold K=2,3
//   B 4x16: v0 holds K=0 (lo half)/K=2 (hi half); v1 holds K=1/K=3, N=lane&15
//   C 16x16: VGPR r <-> row r + 8*(lane>>4), col = lane&15
// ---------------------------------------------------------------------------
template <int K, int NC, int CPW>
__global__ __launch_bounds__(256) void wmma_gemm_f32(const float* __restrict__ A,
                                                     const float* __restrict__ B,
                                                     float* __restrict__ C,
                                                     int M) {
    constexpr int nColT  = NC / 16;
    constexpr int groups = nColT / CPW;   // column groups per row tile
    int wave = threadIdx.x >> 5;
    int lane = threadIdx.x & 31;
    int hl   = lane >> 4;   // half-wave -> K pair selector
    int l    = lane & 15;
    int tile  = blockIdx.x * (blockDim.x >> 5) + wave;
    int total = (M >> 4) * groups;
    if (tile >= total) return;   // wave-uniform exit; EXEC stays all-ones for WMMA
    int m0 = (tile / groups) << 4;
    int n0 = (tile % groups) * (CPW * 16);

    const float* arow = A + (size_t)(m0 + l) * K + 2 * hl;
    const float* bcol = B + (size_t)(2 * hl) * NC + n0 + l;

    v8f acc[CPW];
#pragma unroll
    for (int i = 0; i < CPW; ++i) acc[i] = {};

    for (int k0 = 0; k0 < K; k0 += 4) {
        v2f a;
        a.x = arow[k0 + 0];
        a.y = arow[k0 + 1];
#pragma unroll
        for (int ct = 0; ct < CPW; ++ct) {
            v2f b;
            b.x = bcol[(size_t)k0 * NC + ct * 16];
            b.y = bcol[(size_t)(k0 + 1) * NC + ct * 16];
            acc[ct] = __builtin_amdgcn_wmma_f32_16x16x4_f32(false, a, false, b,
                                                            (short)0, acc[ct], false, false);
        }
    }

#pragma unroll
    for (int ct = 0; ct < CPW; ++ct) {
        float* crow = C + (size_t)(m0 + 8 * hl) * NC + n0 + ct * 16 + l;
#pragma unroll
        for (int r = 0; r < 8; ++r) crow[(size_t)r * NC] = acc[ct][r];
    }
}

// ---------------------------------------------------------------------------
// attention dots, one wave per node (coalesced reads of h):
// asrc[n,h] = <h[n,h,:], att_src[h,:]>, adst likewise
// ---------------------------------------------------------------------------
template <int H, int CH>
__global__ void att_dots_kernel(const float* __restrict__ h,
                                const float* __restrict__ att_s,
                                const float* __restrict__ att_d,
                                float* __restrict__ asrc, float* __restrict__ adst,
                                int n) {
    constexpr int C = H * CH;
    int wave = threadIdx.x >> 5;
    int lane = threadIdx.x & 31;
    int node = blockIdx.x * (blockDim.x >> 5) + wave;
    if (node >= n) return;
    const float* hp = h + (size_t)node * C;

    float s[H], d[H];
#pragma unroll
    for (int i = 0; i < H; ++i) { s[i] = 0.f; d[i] = 0.f; }

#pragma unroll
    for (int i = 0; i < (C + 31) / 32; ++i) {
        int c = lane + i * 32;
        if (c < C) {
            int hd = c / CH;
            float t = hp[c];
            s[hd] += t * att_s[c];
            d[hd] += t * att_d[c];
        }
    }
#pragma unroll
    for (int i = 0; i < H; ++i) {
#pragma unroll
        for (int off = 16; off; off >>= 1) {
            s[i] += __shfl_xor(s[i], off, 32);
            d[i] += __shfl_xor(d[i], off, 32);
        }
        if (lane == 0) {
            asrc[(size_t)node * H + i] = s[i];
            adst[(size_t)node * H + i] = d[i];
        }
    }
}

// ---------------------------------------------------------------------------
// float atomic max via sign-split integer atomics (monotone IEEE mapping)
// ---------------------------------------------------------------------------
__device__ inline void atomicMaxF(float* addr, float v) {
    if (v >= 0.0f)
        atomicMax((int*)addr, __float_as_int(v));
    else
        atomicMin((unsigned int*)addr, (unsigned int)__float_as_int(v));
}

template <int H>
__global__ void edge_max_kernel(const int* __restrict__ src, const int* __restrict__ dst,
                                const float* __restrict__ asrc, const float* __restrict__ adst,
                                float* __restrict__ emax, int E) {
    int e = blockIdx.x * blockDim.x + threadIdx.x;
    if (e >= E) return;
    int s = src[e], d = dst[e];
#pragma unroll
    for (int h = 0; h < H; ++h) {
        float v = asrc[s * H + h] + adst[d * H + h];
        v = (v >= 0.f) ? v : NEG_SLOPE * v;
        atomicMaxF(&emax[d * H + h], v);
    }
}

// ---------------------------------------------------------------------------
// edge aggregation (dominant pass). One lane handles 4 channels of one edge:
//   float4 gather of h[src] (global_load_b128) + 4 atomic f32 adds to out_raw.
//   w = exp(leaky(a_src[s]+a_dst[d]) - emax[d]); denom[d] += w (one lane/head)
// ---------------------------------------------------------------------------
template <int H, int CH>
__global__ __launch_bounds__(256) void edge_agg_kernel(
        const int* __restrict__ src, const int* __restrict__ dst,
        const float* __restrict__ hfeat,
        const float* __restrict__ asrc, const float* __restrict__ adst,
        const float* __restrict__ emax,
        float* __restrict__ denom, float* __restrict__ outraw, int E) {
    constexpr int C   = H * CH;
    constexpr int LPE = C / 4;     // lanes per edge
    constexpr int EPB = 256 / LPE; // edges per block
    int li   = threadIdx.x % LPE;
    int esub = threadIdx.x / LPE;
    int e = blockIdx.x * EPB + esub;
    if (e >= E) return;
    int s = src[e], d = dst[e];
    int c0 = li * 4;
    int h  = c0 / CH;

    float v = asrc[s * H + h] + adst[d * H + h];
    v = (v >= 0.f) ? v : NEG_SLOPE * v;
    float wgt = __expf(v - emax[d * H + h]);
    if ((c0 % CH) == 0) atomicAdd(&denom[d * H + h], wgt);

    const float4 hv = *(const float4*)(hfeat + (size_t)s * C + c0);
    float* op = outraw + (size_t)d * C + c0;
    atomicAdd(op + 0, hv.x * wgt);
    atomicAdd(op + 1, hv.y * wgt);
    atomicAdd(op + 2, hv.z * wgt);
    atomicAdd(op + 3, hv.w * wgt);
}

// ---------------------------------------------------------------------------
// layer-1 finalize: divide by denom, add bias, ELU; writes h_act (aliases h1)
// ---------------------------------------------------------------------------
__global__ void finalize_elu_kernel(const float* __restrict__ outraw,
                                    const float* __restrict__ denom,
                                    const float* __restrict__ bias,
                                    float* __restrict__ hact,
                                    int n, int heads, int ch) {
    size_t i = (size_t)blockIdx.x * blockDim.x + threadIdx.x;
    size_t total = (size_t)n * heads * ch;
    if (i >= total) return;
    int C = heads * ch;
    int node = (int)(i / C);
    int c = (int)(i % C);
    int hd = c / ch;
    float val = outraw[i] / (denom[node * heads + hd] + EPS_) + bias[c];
    hact[i] = (val > 0.f) ? val : expm1f(val);
}

// ---------------------------------------------------------------------------
// layer-2 finalize: divide by denom, add bias, softmax over 16 channels
// ---------------------------------------------------------------------------
__global__ void finalize_softmax_kernel(const float* __restrict__ outraw,
                                        const float* __restrict__ denom,
                                        const float* __restrict__ bias,
                                        float* __restrict__ out, int n) {
    int node = blockIdx.x * blockDim.x + threadIdx.x;
    if (node >= n) return;
    float den = denom[node] + EPS_;
    float v[16];
    float m = -INFINITY;
#pragma unroll
    for (int c = 0; c < 16; ++c) {
        v[c] = outraw[(size_t)node * 16 + c] / den + bias[c];
        m = fmaxf(m, v[c]);
    }
    float sum = 0.f;
#pragma unroll
    for (int c = 0; c < 16; ++c) {
        float t = __expf(v[c] - m);
        v[c] = t;
        sum += t;
    }
    float inv = 1.f / sum;
#pragma unroll
    for (int c = 0; c < 16; ++c) out[(size_t)node * 16 + c] = v[c] * inv;
}

// ---------------------------------------------------------------------------
// launch
// inputs: 0=x 1=W1 2=att_src1 3=att_dst1 4=b1 5=W2 6=att_src2 7=att_dst2
//         8=b2 9=edge_index
// ---------------------------------------------------------------------------
extern "C" void kernel_launch(void* const* d_in, const int* in_sizes, int n_in,
                              void* d_out, int out_size, void* d_ws, size_t ws_size,
                              hipStream_t stream) {
    const float* x   = (const float*)d_in[0];
    const float* W1  = (const float*)d_in[1];
    const float* as1 = (const float*)d_in[2];
    const float* ad1 = (const float*)d_in[3];
    const float* b1  = (const float*)d_in[4];
    const float* W2  = (const float*)d_in[5];
    const float* as2 = (const float*)d_in[6];
    const float* ad2 = (const float*)d_in[7];
    const float* b2  = (const float*)d_in[8];
    const int*   ei  = (const int*)d_in[9];

    const int N = in_sizes[0] / 128;   // 50000
    const int E = in_sizes[9] / 2;     // 1600000
    const int* src = ei;
    const int* dst = ei + E;

    // workspace layout (~60 MB of f32); all offsets 16B-aligned
    float* w = (float*)d_ws;
    float* h1      = w; w += (size_t)N * 128;   // reused as h_act after ELU
    float* outraw1 = w; w += (size_t)N * 128;
    float* asrc1   = w; w += (size_t)N * 2;
    float* adst1   = w; w += (size_t)N * 2;
    float* emax1   = w; w += (size_t)N * 2;
    float* denom1  = w; w += (size_t)N * 2;
    float* h2      = w; w += (size_t)N * 16;
    float* outraw2 = w; w += (size_t)N * 16;
    float* asrc2   = w; w += (size_t)N;
    float* adst2   = w; w += (size_t)N;
    float* emax2   = w; w += (size_t)N;
    float* denom2  = w; w += (size_t)N;

    // re-init accumulators every call (harness does not re-poison)
    fill_kernel<<<1024, 256, 0, stream>>>(outraw1, 0.f, (size_t)N * 128);
    fill_kernel<<<64, 256, 0, stream>>>(denom1, 0.f, (size_t)N * 2);
    fill_kernel<<<64, 256, 0, stream>>>(emax1, -INFINITY, (size_t)N * 2);
    fill_kernel<<<256, 256, 0, stream>>>(outraw2, 0.f, (size_t)N * 16);
    fill_kernel<<<64, 256, 0, stream>>>(denom2, 0.f, (size_t)N);
    fill_kernel<<<64, 256, 0, stream>>>(emax2, -INFINITY, (size_t)N);

    // ---- layer 1: GATConv(128 -> 2x64), concat ----
    {
        int tiles = (N / 16) * 2;   // 2 column groups of 4 tiles (CPW=4)
        wmma_gemm_f32<128, 128, 4><<<cdiv(tiles, 8), 256, 0, stream>>>(x, W1, h1, N);
        att_dots_kernel<2, 64><<<cdiv(N, 8), 256, 0, stream>>>(h1, as1, ad1, asrc1, adst1, N);
        edge_max_kernel<2><<<cdiv(E, 256), 256, 0, stream>>>(src, dst, asrc1, adst1, emax1, E);
        edge_agg_kernel<2, 64><<<cdiv(E, 8), 256, 0, stream>>>(src, dst, h1, asrc1, adst1, emax1,
                                                               denom1, outraw1, E);
        finalize_elu_kernel<<<cdiv((long)N * 128, 256), 256, 0, stream>>>(outraw1, denom1, b1, h1, N, 2, 64);
    }

    // ---- layer 2: GATConv(128 -> 16), heads=1, then softmax ----
    {
        int tiles = (N / 16);       // 1 column group (CPW=1)
        wmma_gemm_f32<128, 16, 1><<<cdiv(tiles, 8), 256, 0, stream>>>(h1, W2, h2, N);
        att_dots_kernel<1, 16><<<cdiv(N, 8), 256, 0, stream>>>(h2, as2, ad2, asrc2, adst2, N);
        edge_max_kernel<1><<<cdiv(E, 256), 256, 0, stream>>>(src, dst, asrc2, adst2, emax2, E);
        edge_agg_kernel<1, 16><<<cdiv(E, 64), 256, 0, stream>>>(src, dst, h2, asrc2, adst2, emax2,
                                                                denom2, outraw2, E);
        finalize_softmax_kernel<<<cdiv(N, 256), 256, 0, stream>>>(outraw2, denom2, b2, (float*)d_out, N);
    }
}